// SpectralTemporalConv2d_74663711474425
// MI455X (gfx1250) — compile-verified
//
#include <hip/hip_runtime.h>
#include <hip/hip_bf16.h>

// ---------------------------------------------------------------------------
// SpectralTemporalConv2d for MI455X (gfx1250, wave32, WMMA, async-LDS)
//
// B=16, CIN=COUT=32, H=W=64, M1=M2=12, TDIM=THID=256
// Dominant cost: streaming 604 MB of complex64 weights (w1,w2) from HBM
// (~26 us floor at 23.3 TB/s). All matrix math on the weight stream uses
// V_WMMA_F32_16X16X4_F32 (full f32, still memory-bound: 4.8 GFLOP needs only
// ~185 TF/s to keep up with the stream). LDS staging uses
// GLOBAL_LOAD_ASYNC_TO_LDS_B128 tracked by ASYNCcnt.
// ---------------------------------------------------------------------------

typedef float v2f __attribute__((ext_vector_type(2)));
typedef float v8f __attribute__((ext_vector_type(8)));

#define WMMA_F32_4(a, b, c)                                                    \
    __builtin_amdgcn_wmma_f32_16x16x4_f32(false, (a), false, (b), (short)0,    \
                                          (c), false, false)

// Async global->LDS copy of 16 bytes (CDNA5, tracked with ASYNCcnt).
__device__ __forceinline__ void async_copy_b128(unsigned lds_addr,
                                                const void* gaddr) {
  asm volatile("global_load_async_to_lds_b128 %0, %1, off"
               :: "v"(lds_addr), "v"(gaddr)
               : "memory");
}
__device__ __forceinline__ void wait_async0() {
  asm volatile("s_wait_asynccnt 0x0" ::: "memory");
}

// ---------------------------------------------------------------------------
// Kernel 1: one time-MLP layer: out[16x256] = relu(act[16xK] @ w[Kx256] + b)
// One wave (32 threads) per 16x16 output tile, K-loop in steps of 4 using
// f32 WMMA. A-matrix lanes: m = lane%16, K-pair = (lane<16 ? 0,1 : 2,3).
// ---------------------------------------------------------------------------
__global__ __launch_bounds__(32) void mlp_layer_kernel(
    const float* __restrict__ act, const float* __restrict__ wmat,
    const float* __restrict__ bias, float* __restrict__ out, int K) {
  int lane = threadIdx.x & 31;
  int m    = lane & 15;
  int k0   = (lane >> 4) << 1;                 // 0 or 2
  int col  = blockIdx.x * 16 + (lane & 15);    // output column (N)
  const float* ap = act + m * K + k0;
  v8f acc = {};
  for (int t0 = 0; t0 < K; t0 += 4) {
    v2f a; a[0] = ap[t0];                      a[1] = ap[t0 + 1];
    v2f b; b[0] = wmat[(t0 + k0) * 256 + col]; b[1] = wmat[(t0 + k0 + 1) * 256 + col];
    acc = WMMA_F32_4(a, b, acc);
  }
  float bb = bias[col];
#pragma unroll
  for (int j = 0; j < 8; ++j) {
    int mo  = j + ((lane >> 4) << 3);          // D row: lanes16-31 hold M=8..15
    float v = acc[j] + bb;
    out[mo * 256 + col] = v > 0.f ? v : 0.f;
  }
}

// ---------------------------------------------------------------------------
// Kernel 2: mode-restricted forward DFT. One block per (b,i) image.
// Produces xf[(half*144 + ky*12 + kx)*16 + b][i]  (288 modes per image).
// Only 24x12 of the 64x33 rfft2 output is ever used -> direct 2-pass DFT.
// ---------------------------------------------------------------------------
__global__ __launch_bounds__(256) void fft_modes_kernel(
    const float* __restrict__ x, float2* __restrict__ xf) {
  __shared__ float  xt[64 * 64];
  __shared__ float2 tw[64];          // e^{+2*pi*i*k/64}
  __shared__ float2 T[64 * 12];      // after x-direction pass: [y][kx]
  int tid = threadIdx.x;
  int b = blockIdx.x >> 5;
  int i = blockIdx.x & 31;
  const float* xp = x + (size_t)(b * 32 + i) * 4096;
  // async-stage the 16KB image tile (1024 16B chunks, 4 per thread)
  {
    unsigned xt_base = (unsigned)(size_t)(&xt[0]);
#pragma unroll
    for (int k = 0; k < 4; ++k) {
      int c = tid + k * 256;
      async_copy_b128(xt_base + (unsigned)c * 16u, xp + c * 4);
    }
  }
  if (tid < 64) {
    float s, c;
    __sincosf(6.2831853071795864f * (float)tid / 64.f, &s, &c);
    tw[tid] = make_float2(c, s);
  }
  wait_async0();
  __syncthreads();
  // pass 1 along W: T[y][kx] = sum_x xt[y][x] * e^{-2pi i x kx/64}
  for (int idx = tid; idx < 768; idx += 256) {
    int y = idx / 12, kx = idx % 12;
    float tr = 0.f, ti = 0.f;
    for (int xx = 0; xx < 64; ++xx) {
      float  v = xt[y * 64 + xx];
      float2 w = tw[(xx * kx) & 63];
      tr += v * w.x;
      ti -= v * w.y;
    }
    T[idx] = make_float2(tr, ti);
  }
  __syncthreads();
  // pass 2 along H: keep only ky in {0..11} (half 0) and {52..63} (half 1)
  for (int idx = tid; idx < 288; idx += 256) {
    int half = idx / 144;
    int r    = idx % 144;
    int kyI  = r / 12, kx = r % 12;
    int ky   = half ? (52 + kyI) : kyI;
    float fr = 0.f, fi = 0.f;
    for (int y = 0; y < 64; ++y) {
      float2 t = T[y * 12 + kx];
      float2 w = tw[(y * ky) & 63];
      fr += t.x * w.x + t.y * w.y;   // t * conj(tw) = e^{-i...}
      fi += t.y * w.x - t.x * w.y;
    }
    xf[((size_t)idx * 16 + b) * 32 + i] = make_float2(fr, fi);
  }
}

// ---------------------------------------------------------------------------
// Kernel 3 (hot): fused spectral conv.
// Grid = half(2) x mode(144) x o-tile(4).  Block = 256 threads = 8 waves.
// Step A (WMMA): U[b, io] = sum_t h[b,t] * w[io, t]   (re & im separately)
//   A = h tile from padded LDS (16x256, +4 pad kills bank conflicts),
//   B = per-lane global_load_b128 of 2 consecutive complex weights
//       (each lane owns one io column and walks its contiguous 2KB t-run,
//        so every HBM cache line is consumed exactly once).
// Step B (LDS): out[b,o] = sum_i xf[b,i] * U[b, i*8+o]   (complex).
// Each block owns disjoint outputs -> plain global stores, no atomics.
// h and xf staged with GLOBAL_LOAD_ASYNC_TO_LDS_B128 (ASYNCcnt).
// ---------------------------------------------------------------------------
__global__ __launch_bounds__(256) void spectral_kernel(
    const float* __restrict__ hmat,   // 16 x 256 (time-MLP output)
    const float2* __restrict__ xf,    // staged modes [(half,mode)][b][i]
    const float* __restrict__ w1, const float* __restrict__ w2,
    float2* __restrict__ outft) {     // [b][o][half][mx][my]
  __shared__ float  hl[16 * 260];     // padded h tile
  __shared__ float2 xfl[512];         // xf slice [b][i]
  __shared__ float2 Ul[256 * 17];     // U[io][b] (padded)
  int tid  = threadIdx.x;
  int lane = tid & 31;
  int wave = tid >> 5;

  int blk  = blockIdx.x;
  int half = blk / 576;
  int rem  = blk % 576;
  int mode = rem >> 2;                // mx*12 + my
  int ot   = rem & 3;                 // 8 output channels per tile

  // async-stage h (16KB; 1024 16B chunks, rows of 256 floats = 64 clean
  // chunks each, so no chunk straddles the +4 row padding) and the xf slice
  {
    unsigned hl_base  = (unsigned)(size_t)(&hl[0]);
    unsigned xfl_base = (unsigned)(size_t)(&xfl[0]);
#pragma unroll
    for (int k = 0; k < 4; ++k) {
      int c   = tid + k * 256;        // chunk id 0..1023
      int row = c >> 6, col4 = c & 63;
      async_copy_b128(hl_base + (unsigned)(row * 260 + col4 * 4) * 4u,
                      hmat + row * 256 + col4 * 4);
    }
    const float2* src = xf + (size_t)(half * 144 + mode) * 512;
    async_copy_b128(xfl_base + (unsigned)tid * 16u, src + tid * 2);
  }
  wait_async0();
  __syncthreads();

  const float* wsel = half ? w2 : w1;
  int n     = lane & 15;              // B-matrix column within group
  int k0    = (lane >> 4) << 1;       // t sub-offset: 0 or 2
  int hbase = (lane & 15) * 260 + k0; // A-matrix row base in LDS

  // each wave handles two 16-column io groups (16 groups cover io=0..255)
  const float *wp0, *wp1;
  {
    int io0 = (wave * 2) * 16 + n;
    int io1 = (wave * 2 + 1) * 16 + n;
    int i0 = io0 >> 3, o0 = ot * 8 + (io0 & 7);
    int i1 = io1 >> 3, o1 = ot * 8 + (io1 & 7);
    wp0 = wsel + ((size_t)(i0 * 32 + o0) * 144 + mode) * 512 + k0 * 2;
    wp1 = wsel + ((size_t)(i1 * 32 + o1) * 144 + mode) * 512 + k0 * 2;
  }

  v8f ar0 = {}, ai0 = {}, ar1 = {}, ai1 = {};
  for (int t0 = 0; t0 < 256; t0 += 4) {
    if ((t0 & 31) == 0) {             // ride 512B ahead of the weight stream
      __builtin_prefetch(wp0 + t0 * 2 + 128, 0, 0);
      __builtin_prefetch(wp1 + t0 * 2 + 128, 0, 0);
    }
    v2f a; a[0] = hl[hbase + t0]; a[1] = hl[hbase + t0 + 1];
    float4 wv0 = *(const float4*)(wp0 + t0 * 2);  // re,im,re,im (2 complex t)
    float4 wv1 = *(const float4*)(wp1 + t0 * 2);
    v2f br0 = {wv0.x, wv0.z}, bi0 = {wv0.y, wv0.w};
    v2f br1 = {wv1.x, wv1.z}, bi1 = {wv1.y, wv1.w};
    ar0 = WMMA_F32_4(a, br0, ar0);
    ai0 = WMMA_F32_4(a, bi0, ai0);
    ar1 = WMMA_F32_4(a, br1, ar1);
    ai1 = WMMA_F32_4(a, bi1, ai1);
  }

  // scatter U tiles to LDS: Ul[io][b]
  int mofs = (lane >> 4) << 3;
#pragma unroll
  for (int j = 0; j < 8; ++j) {
    int m = j + mofs;
    Ul[((wave * 2) * 16 + n) * 17 + m]     = make_float2(ar0[j], ai0[j]);
    Ul[((wave * 2 + 1) * 16 + n) * 17 + m] = make_float2(ar1[j], ai1[j]);
  }
  __syncthreads();

  // contraction over i: out[b, o] = sum_i xf[b,i] * U[b, i*8+o]
  if (tid < 128) {
    int b = tid & 15, o = tid >> 4;
    float sr = 0.f, si = 0.f;
    for (int i = 0; i < 32; ++i) {
      float2 xv = xfl[b * 32 + i];
      float2 uv = Ul[(i * 8 + o) * 17 + b];
      sr += xv.x * uv.x - xv.y * uv.y;
      si += xv.x * uv.y + xv.y * uv.x;
    }
    int og = ot * 8 + o;
    outft[(size_t)(b * 32 + og) * 288 + half * 144 + mode] =
        make_float2(sr, si);
  }
}

// ---------------------------------------------------------------------------
// Kernel 4: sparse inverse transform. One block per (b,o) image.
// ifft along H over the 24 nonzero ky, then Hermitian irfft along W over
// the 12 nonzero kx (coef 0 weight 1, others weight 2), scaled by 1/4096.
// ---------------------------------------------------------------------------
__global__ __launch_bounds__(256) void ifft_kernel(
    const float2* __restrict__ outft, float* __restrict__ out) {
  __shared__ float2 tw[64];          // e^{+2*pi*i*k/64}
  __shared__ float2 F[288];
  __shared__ float2 S[64 * 12];      // [y][kx]
  int tid = threadIdx.x;
  int bo  = blockIdx.x;
  const float2* fp = outft + (size_t)bo * 288;
  if (tid < 64) {
    float s, c;
    __sincosf(6.2831853071795864f * (float)tid / 64.f, &s, &c);
    tw[tid] = make_float2(c, s);
  }
  // async-stage the 288 modes (2304B = 144 16B chunks)
  if (tid < 144) {
    unsigned f_base = (unsigned)(size_t)(&F[0]);
    async_copy_b128(f_base + (unsigned)tid * 16u, fp + tid * 2);
  }
  wait_async0();
  __syncthreads();
  // inverse along H (only 24 nonzero rows)
  for (int idx = tid; idx < 768; idx += 256) {
    int y = idx / 12, kx = idx % 12;
    float sr = 0.f, si = 0.f;
#pragma unroll
    for (int h = 0; h < 2; ++h)
      for (int kyI = 0; kyI < 12; ++kyI) {
        int   ky = h ? (52 + kyI) : kyI;
        float2 f = F[h * 144 + kyI * 12 + kx];
        float2 w = tw[(y * ky) & 63];
        sr += f.x * w.x - f.y * w.y;
        si += f.x * w.y + f.y * w.x;
      }
    S[idx] = make_float2(sr, si);
  }
  __syncthreads();
  // Hermitian inverse along W
  float* op = out + (size_t)bo * 4096;
  for (int idx = tid; idx < 4096; idx += 256) {
    int y = idx >> 6, xx = idx & 63;
    float acc = 0.f;
#pragma unroll
    for (int kx = 0; kx < 12; ++kx) {
      float2 s  = S[y * 12 + kx];
      float2 w  = tw[(xx * kx) & 63];
      float  re = s.x * w.x - s.y * w.y;
      acc += (kx == 0) ? re : 2.f * re;
    }
    op[idx] = acc * (1.f / 4096.f);
  }
}

// ---------------------------------------------------------------------------
extern "C" void kernel_launch(void* const* d_in, const int* in_sizes, int n_in,
                              void* d_out, int out_size, void* d_ws,
                              size_t ws_size, hipStream_t stream) {
  (void)in_sizes; (void)n_in; (void)out_size; (void)ws_size;
  const float* t   = (const float*)d_in[0];
  const float* x   = (const float*)d_in[1];
  const float* w1  = (const float*)d_in[2];   // complex64 -> interleaved f32
  const float* w2  = (const float*)d_in[3];
  const float* mw0 = (const float*)d_in[4];
  const float* mb0 = (const float*)d_in[5];
  const float* mw1 = (const float*)d_in[6];
  const float* mb1 = (const float*)d_in[7];
  const float* mw2 = (const float*)d_in[8];
  const float* mb2 = (const float*)d_in[9];
  float* out = (float*)d_out;
  float* ws  = (float*)d_ws;

  // workspace layout (floats): h0 | h1 | h2 | xf (294912) | outft (294912)
  float*  h0  = ws;
  float*  h1  = ws + 4096;
  float*  h2  = ws + 8192;
  float2* xf  = (float2*)(ws + 12288);
  float2* oft = (float2*)(ws + 12288 + 294912);

  mlp_layer_kernel<<<16, 32, 0, stream>>>(t,  mw0, mb0, h0, 32);
  mlp_layer_kernel<<<16, 32, 0, stream>>>(h0, mw1, mb1, h1, 256);
  mlp_layer_kernel<<<16, 32, 0, stream>>>(h1, mw2, mb2, h2, 256);
  fft_modes_kernel<<<512, 256, 0, stream>>>(x, xf);
  spectral_kernel<<<1152, 256, 0, stream>>>(h2, xf, w1, w2, oft);
  ifft_kernel<<<512, 256, 0, stream>>>(oft, out);
}